// OHNMLoss_72430328480988
// MI455X (gfx1250) — compile-verified
//
#include <hip/hip_runtime.h>
#include <stdint.h>

// ---------------------------------------------------------------------------
// OHNM loss for MI455X (gfx1250, wave32).
// Strategy: 2-level radix-select on monotonic float keys + streaming BCE sums.
//   K0: zero scratch histograms/scalars
//   K1: LDS-privatized 32768-bin histogram of negatives (128KB LDS, CDNA5 big
//       LDS), positive BCE partial sums, ballot-built positive bitmask
//   K2: suffix-scan histogram -> threshold bucket B + count_above
//   K3: second data pass staged through LDS with TDM tensor_load_to_lds
//       (double buffered, s_wait_tensorcnt), BCE sum above bucket B,
//       131072-bin low-bit histogram inside bucket B
//   K4: scan low histogram -> exact k-th key; finalize scalar loss
// Block reductions use V_WMMA_F32_16X16X4_F32 (B = ones) as an f32-exact
// 64-way add tree step.
// ---------------------------------------------------------------------------

#define TOT        16777216   // 32 * 524288
#define NB1        32768      // top 15 key bits
#define NB2        131072     // low 17 key bits
#define K1_BLOCKS  512
#define K3_BLOCKS  512
#define TILE       8192       // floats per TDM tile (32 KB)

// scratch layout (uint32 units)
#define H1_OFF     0u
#define H2_OFF     32768u
#define PSUM_OFF   163840u    // 512 float partials
#define PCNT_OFF   164352u    // 512 float partials
#define HSUM_OFF   164864u    // 512 float partials
#define SCAL_OFF   165376u    // [0]=B [1]=count_above [2]=pos_sum(f32) [3]=pos_cnt(f32)
#define MASK_OFF   165888u    // 524288 words of positive bitmask
#define ZERO_WORDS 165888u

typedef __attribute__((ext_vector_type(2))) float    v2f;
typedef __attribute__((ext_vector_type(8))) float    v8f;
typedef __attribute__((ext_vector_type(4))) unsigned tdm_v4u;
typedef __attribute__((ext_vector_type(8))) int      tdm_v8i;
typedef __attribute__((ext_vector_type(4))) int      tdm_v4i;

// order-preserving float -> uint key
__device__ inline unsigned mono(float f) {
  unsigned u = __float_as_uint(f);
  return (u & 0x80000000u) ? ~u : (u | 0x80000000u);
}
__device__ inline float unmono(unsigned k) {
  return (k & 0x80000000u) ? __uint_as_float(k ^ 0x80000000u)
                           : __uint_as_float(~k);
}
// stable log(1 + e^v)  == BCE-with-logits at label 0
__device__ inline float bce0(float v) {
  return fmaxf(v, 0.0f) + log1pf(expf(-fabsf(v)));
}

// Wave32 sum via V_WMMA_F32_16X16X4_F32: A(16x4) vgpr0 = per-lane value,
// B(4x16) = ones -> D[m][n] = v(m)+v(m+16). Lane0 holds rows 0-7 in d0..d7,
// lane16 holds rows 8-15. Valid result on lane 0 only. EXEC must be all-1s.
__device__ inline float wmmaWaveSum(float v) {
  v2f a; a[0] = v;    a[1] = 0.0f;
  v2f b; b[0] = 1.0f; b[1] = 1.0f;
  v8f c = {0.f,0.f,0.f,0.f,0.f,0.f,0.f,0.f};
  c = __builtin_amdgcn_wmma_f32_16x16x4_f32(false, a, false, b,
                                            (short)0, c, false, false);
  float s = c[0]+c[1]+c[2]+c[3]+c[4]+c[5]+c[6]+c[7];
  s += __shfl(s, 16, 32);   // lane0: rows0-7 + rows8-15 = full wave sum
  return s;
}

// TDM: DMA one 32KB tile of x into LDS (1-D tensor descriptor, data_size=4B)
__device__ inline void tdmLoadTile(const float* x, int tileIdx, unsigned ldsByteOff) {
  unsigned long long ga =
      (unsigned long long)(uintptr_t)(x + (size_t)tileIdx * TILE);
  tdm_v4u g0;
  g0[0] = 1u;                                   // count=1, user descriptor
  g0[1] = ldsByteOff;                           // lds_addr
  g0[2] = (unsigned)(ga & 0xffffffffull);       // global_addr[31:0]
  g0[3] = ((unsigned)(ga >> 32) & 0x01ffffffu)  // global_addr[56:32]
          | 0x80000000u;                        // type=2 ("image")
  tdm_v8i g1;
  g1[0] = (int)(2u << 16);                      // data_size = 4 bytes
  g1[1] = (int)((unsigned)TILE << 16);          // tensor_dim0[15:0]
  g1[2] = 0;                                    // tensor_dim0[31:16], dim1=0
  g1[3] = (int)((unsigned)TILE << 16);          // tile_dim0 = TILE
  g1[4] = 0;                                    // tile_dim1/2 unused
  g1[5] = (int)TILE;                            // tensor_dim0_stride
  g1[6] = 0; g1[7] = 0;
  tdm_v4i g2 = (tdm_v4i)0;
  tdm_v4i g3 = (tdm_v4i)0;
#if __has_include(<hip/amd_detail/amd_gfx1250_TDM.h>)
  tdm_v8i g4 = (tdm_v8i)0;
  __builtin_amdgcn_tensor_load_to_lds(g0, g1, g2, g3, g4, 0);
#else
  __builtin_amdgcn_tensor_load_to_lds(g0, g1, g2, g3, 0);
#endif
}

// ---------------------------------------------------------------- K0: init
__global__ void k0_init(unsigned* __restrict__ ws) {
  unsigned i = blockIdx.x * blockDim.x + threadIdx.x;
  if (i < ZERO_WORDS) ws[i] = 0u;
}

// ------------------- K1: histogram + positive sums + positive bitmask -----
__global__ void __launch_bounds__(1024)
k1_hist(const float* __restrict__ x, const float* __restrict__ y,
        unsigned* __restrict__ ws) {
  __shared__ unsigned lh[NB1];    // 128 KB LDS histogram (CDNA5 big LDS)
  __shared__ float    lred[32];
  const int tid  = threadIdx.x;
  const int lane = tid & 31;
  const int wid  = tid >> 5;
  for (int i = tid; i < NB1; i += 1024) lh[i] = 0u;
  __syncthreads();

  unsigned* maskW = ws + MASK_OFF;
  float ps = 0.0f, pc = 0.0f;
  const int gw0 = blockIdx.x * 32 + wid;      // global wave id, 0..16383
  for (int it = 0; it < 8; ++it) {
    const int eBase = (gw0 + it * 16384) * 128;  // wave handles 128 contig elems
    const float* xp = x + eBase + lane;
    const float* yp = y + eBase + lane;
    unsigned bal[4];
#pragma unroll
    for (int t = 0; t < 4; ++t) {
      float xv = xp[32 * t];
      float yv = yp[32 * t];
      bool pos = yv > 0.0f;
      bal[t] = (unsigned)__ballot(pos);        // wave32: low 32 bits
      if (pos) {
        ps += bce0(xv) - xv * yv;
        pc += 1.0f;
      } else {
        unsigned key = mono(xv);
        atomicAdd(&lh[key >> 17], 1u);         // ds_add, LDS-privatized
      }
    }
    unsigned sel = (lane == 0) ? bal[0] : (lane == 1) ? bal[1]
                 : (lane == 2) ? bal[2] : bal[3];
    if (lane < 4) maskW[(eBase >> 5) + lane] = sel;
  }
  __syncthreads();
  // merge LDS histogram into global (sparse: most bins are zero)
  for (int i = tid; i < NB1; i += 1024) {
    unsigned c = lh[i];
    if (c) atomicAdd(ws + H1_OFF + i, c);
  }
  // deterministic per-block partial reductions via WMMA tree
  float s = wmmaWaveSum(ps);
  if (lane == 0) lred[wid] = s;
  __syncthreads();
  if (wid == 0) {
    float r = wmmaWaveSum(lred[lane]);
    if (lane == 0) ((float*)(ws + PSUM_OFF))[blockIdx.x] = r;
  }
  __syncthreads();
  s = wmmaWaveSum(pc);
  if (lane == 0) lred[wid] = s;
  __syncthreads();
  if (wid == 0) {
    float r = wmmaWaveSum(lred[lane]);
    if (lane == 0) ((float*)(ws + PCNT_OFF))[blockIdx.x] = r;
  }
}

// ------------------- K2: scan histogram -> bucket B, reduce pos partials ---
__global__ void __launch_bounds__(1024)
k2_scan(unsigned* __restrict__ ws, const int* __restrict__ negnum) {
  __shared__ unsigned sc[1024];
  __shared__ float    sf[1024];
  const int tid = threadIdx.x;

  float a = (tid < 512) ? ((const float*)(ws + PSUM_OFF))[tid] : 0.0f;
  float b = (tid < 512) ? ((const float*)(ws + PCNT_OFF))[tid] : 0.0f;
  sf[tid] = a; __syncthreads();
  for (int off = 512; off > 0; off >>= 1) {
    if (tid < off) sf[tid] += sf[tid + off];
    __syncthreads();
  }
  if (tid == 0) ((float*)(ws + SCAL_OFF))[2] = sf[0];
  __syncthreads();
  sf[tid] = b; __syncthreads();
  for (int off = 512; off > 0; off >>= 1) {
    if (tid < off) sf[tid] += sf[tid + off];
    __syncthreads();
  }
  if (tid == 0) ((float*)(ws + SCAL_OFF))[3] = sf[0];

  // descending suffix-scan: thread i owns bins [NB1-32(i+1), NB1-32i)
  const unsigned k = (unsigned)negnum[0];
  const int hi = NB1 - 32 * tid, lo = hi - 32;
  unsigned csum = 0;
  for (int bin = hi - 1; bin >= lo; --bin) csum += ws[H1_OFF + bin];
  sc[tid] = csum; __syncthreads();
  for (int off = 1; off < 1024; off <<= 1) {
    unsigned t = (tid >= off) ? sc[tid - off] : 0u;
    __syncthreads();
    sc[tid] += t;
    __syncthreads();
  }
  unsigned run = sc[tid] - csum;               // elements strictly above chunk
  for (int bin = hi - 1; bin >= lo; --bin) {
    unsigned c = ws[H1_OFF + bin];
    if (run < k && run + c >= k) {             // k-th element lives here
      ws[SCAL_OFF + 0] = (unsigned)bin;
      ws[SCAL_OFF + 1] = run;
    }
    run += c;
  }
}

// ------------------- K3: TDM-staged second pass ---------------------------
__global__ void __launch_bounds__(256)
k3_pass2(const float* __restrict__ x, unsigned* __restrict__ ws) {
  __shared__ __align__(16) float buf[2][TILE];   // 64 KB double buffer
  __shared__ float lred[8];
  const int tid  = threadIdx.x;
  const int lane = tid & 31;
  const int wid  = tid >> 5;
  const unsigned Bq = ws[SCAL_OFF + 0];
  const unsigned* maskW = ws + MASK_OFF;
  unsigned* h2 = ws + H2_OFF;
  float hs = 0.0f;
  const int tile0 = blockIdx.x * 4;

  if (wid == 0)   // one TDM issue per workgroup (EXEC-independent DMA)
    tdmLoadTile(x, tile0, (unsigned)(uintptr_t)&buf[0][0]);

  for (int t = 0; t < 4; ++t) {
    if (wid == 0) {
      if (t < 3) {
        tdmLoadTile(x, tile0 + t + 1, (unsigned)(uintptr_t)&buf[(t + 1) & 1][0]);
        __builtin_amdgcn_s_wait_tensorcnt(1);   // tile t landed, prefetch flies
      } else {
        __builtin_amdgcn_s_wait_tensorcnt(0);
      }
    }
    __syncthreads();
    const float4* tb = (const float4*)&buf[t & 1][0];
    const int f4base = (tile0 + t) * (TILE / 4);
#pragma unroll
    for (int j = 0; j < 8; ++j) {
      const int f4 = tid + j * 256;
      float4 v = tb[f4];                         // ds_load_b128
      const int gf4 = f4base + f4;
      unsigned mw  = maskW[gf4 >> 3];
      unsigned nib = (mw >> ((gf4 & 7) * 4)) & 0xFu;
      float vv[4] = {v.x, v.y, v.z, v.w};
#pragma unroll
      for (int c = 0; c < 4; ++c) {
        if (!((nib >> c) & 1u)) {                // negative element
          unsigned key = mono(vv[c]);
          unsigned hi17 = key >> 17;
          if (hi17 > Bq) hs += bce0(vv[c]);
          else if (hi17 == Bq) atomicAdd(&h2[key & 0x1FFFFu], 1u);
        }
      }
    }
    __syncthreads();   // all readers done before next overwrite of this buffer
  }
  float s = wmmaWaveSum(hs);
  if (lane == 0) lred[wid] = s;
  __syncthreads();
  if (wid == 0) {
    float v = (lane < 8) ? lred[lane] : 0.0f;
    float r = wmmaWaveSum(v);
    if (lane == 0) ((float*)(ws + HSUM_OFF))[blockIdx.x] = r;
  }
}

// ------------------- K4: exact threshold + finalize -----------------------
__global__ void __launch_bounds__(1024)
k4_final(unsigned* __restrict__ ws, const int* __restrict__ negnum,
         float* __restrict__ out) {
  __shared__ unsigned sc[1024];
  __shared__ float    lred[32];
  __shared__ unsigned sCross, sR2;
  const int tid = threadIdx.x;
  const int lane = tid & 31;
  const int wid  = tid >> 5;
  if (tid == 0) { sCross = 0x7fffffffu; sR2 = 0u; }

  const unsigned Bq    = ws[SCAL_OFF + 0];
  const unsigned above = ws[SCAL_OFF + 1];
  const unsigned k     = (unsigned)negnum[0];
  const unsigned r1    = k - above;            // elements taken from bucket B

  const int hi = NB2 - 128 * tid, lo = hi - 128;
  unsigned csum = 0;
  for (int b = hi - 1; b >= lo; --b) csum += ws[H2_OFF + b];
  sc[tid] = csum; __syncthreads();
  for (int off = 1; off < 1024; off <<= 1) {
    unsigned t = (tid >= off) ? sc[tid - off] : 0u;
    __syncthreads();
    sc[tid] += t;
    __syncthreads();
  }
  unsigned run = sc[tid] - csum;
  for (int b = hi - 1; b >= lo; --b) {
    unsigned c = ws[H2_OFF + b];
    if (run < r1 && run + c >= r1) { sCross = (unsigned)b; sR2 = r1 - run; }
    run += c;
  }
  __syncthreads();
  const unsigned cross = sCross, r2 = sR2;

  // all elements in one 32-bit-key bin share the exact float value
  float w = 0.0f;
  for (int b = hi - 1; b >= lo; --b) {
    if ((unsigned)b > cross) {
      unsigned c = ws[H2_OFF + b];
      if (c) w += (float)c * bce0(unmono((Bq << 17) | (unsigned)b));
    }
  }
  if (tid < 512) w += ((const float*)(ws + HSUM_OFF))[tid];

  float s = wmmaWaveSum(w);
  if (lane == 0) lred[wid] = s;
  __syncthreads();
  if (wid == 0) {
    float r = wmmaWaveSum(lred[lane]);
    if (lane == 0) {
      float tie = (r2 > 0 && cross < NB2)
                    ? (float)r2 * bce0(unmono((Bq << 17) | cross)) : 0.0f;
      float possum = ((const float*)(ws + SCAL_OFF))[2];
      float poscnt = ((const float*)(ws + SCAL_OFF))[3];
      out[0] = (possum + r + tie) / (poscnt + (float)k);
    }
  }
}

// ---------------------------------------------------------------------------
extern "C" void kernel_launch(void* const* d_in, const int* in_sizes, int n_in,
                              void* d_out, int out_size, void* d_ws, size_t ws_size,
                              hipStream_t stream) {
  const float* x = (const float*)d_in[0];
  const float* y = (const float*)d_in[1];
  const int* negnum = (const int*)d_in[2];
  float* out = (float*)d_out;
  unsigned* ws = (unsigned*)d_ws;   // needs ~2.7 MB of scratch

  k0_init <<<(ZERO_WORDS + 255) / 256, 256, 0, stream>>>(ws);
  k1_hist <<<K1_BLOCKS, 1024, 0, stream>>>(x, y, ws);
  k2_scan <<<1, 1024, 0, stream>>>(ws, negnum);
  k3_pass2<<<K3_BLOCKS, 256, 0, stream>>>(x, ws);
  k4_final<<<1, 1024, 0, stream>>>(ws, negnum, out);
}